// PostProcessor_54374285967910
// MI455X (gfx1250) — compile-verified
//
#include <hip/hip_runtime.h>

#define CNUM     81
#define REG      405            // CNUM*5
#define OUTW     486            // CNUM + REG
#define SLABF    512            // padded floats per wave slab (13*32 = 416 rounded up)
#define WAVES_PB 8
#define XCLIP    4.135166556742356f   // log(1000/16)
#define R2D      57.29577951308232f   // 180/pi

__device__ __forceinline__ float wave_max32(float v) {
#pragma unroll
    for (int off = 16; off > 0; off >>= 1)
        v = fmaxf(v, __shfl_xor(v, off, 32));
    return v;
}
__device__ __forceinline__ float wave_sum32(float v) {
#pragma unroll
    for (int off = 16; off > 0; off >>= 1)
        v += __shfl_xor(v, off, 32);
    return v;
}

__global__ __launch_bounds__(256) void
PostProcessor_54374285967910_kernel(const float* __restrict__ logits,
                                    const float* __restrict__ regression,
                                    const float* __restrict__ boxes,
                                    float* __restrict__ out,
                                    int n)
{
    __shared__ float slab[WAVES_PB * SLABF];

    const int lane = threadIdx.x & 31;
    const int wave = threadIdx.x >> 5;
    const int row  = blockIdx.x * WAVES_PB + wave;
    if (row >= n) return;

    // LDS byte offset of this wave's slab: low 32 bits of the generic pointer
    // to __shared__ data ARE the workgroup-relative LDS byte address.
    const unsigned lds_base =
        (unsigned)(unsigned long long)(uintptr_t)(&slab[wave * SLABF]);

    // ---- 1) async-stage the 405-float regression row into LDS (ASYNCcnt) ----
    const float* rrow = regression + (size_t)row * REG;
#pragma unroll
    for (int t = 0; t < 13; ++t) {
        int f  = t * 32 + lane;
        int gf = (f < REG) ? f : 0;           // clamp tail lanes (dup, harmless)
        unsigned ldsa = lds_base + (unsigned)f * 4u;
        const float* ga = rrow + gf;
        asm volatile("global_load_async_to_lds_b32 %0, %1, off"
                     :: "v"(ldsa), "v"(ga) : "memory");
    }

    // ---- 2) softmax over 81 logits (overlaps with the async copy) ----
    const float* lrow = logits + (size_t)row * CNUM;
    float x0 = lrow[lane];
    float x1 = lrow[lane + 32];
    float x2 = (lane < CNUM - 64) ? lrow[lane + 64] : -__builtin_inff();
    float m  = wave_max32(fmaxf(fmaxf(x0, x1), x2));
    float e0 = __expf(x0 - m);
    float e1 = __expf(x1 - m);
    float e2 = (lane < CNUM - 64) ? __expf(x2 - m) : 0.0f;
    float s  = wave_sum32(e0 + e1 + e2);
    float inv = __builtin_amdgcn_rcpf(s);

    float* orow = out + (size_t)row * OUTW;
    orow[lane]      = e0 * inv;
    orow[lane + 32] = e1 * inv;
    if (lane < CNUM - 64) orow[lane + 64] = e2 * inv;

    // ---- 3) proposal box (wave-uniform; L0/L2 broadcast) ----
    const float* brow = boxes + (size_t)row * 5;
    float cx = brow[0], cy = brow[1], bw = brow[2], bh = brow[3], ba = brow[4];

    // ---- 4) wait for the async stage, then decode 405 codes from LDS ----
    asm volatile("s_wait_asynccnt 0" ::: "memory");
    const float* sreg = &slab[wave * SLABF];
#pragma unroll
    for (int t = 0; t < 13; ++t) {
        int j = t * 32 + lane;
        if (j < REG) {
            int   k    = j - (j / 5) * 5;                 // 0..4
            float rel  = sreg[j];
            float invw = (k < 2) ? 0.1f : ((k < 4) ? 0.2f : 1.0f);
            float d    = rel * invw;
            float dim  = ((k == 0) || (k == 2)) ? bw : bh;
            float ctr  = (k == 0) ? cx : cy;
            float lin  = fminf(fmaxf(__builtin_fmaf(d, dim, ctr), 0.0f), 1023.0f);
            float ex   = __expf(fminf(d, XCLIP)) * dim;
            float ang  = __builtin_fmaf(d, R2D, ba);
            orow[CNUM + j] = (k < 2) ? lin : ((k < 4) ? ex : ang);
        }
    }
}

extern "C" void kernel_launch(void* const* d_in, const int* in_sizes, int n_in,
                              void* d_out, int out_size, void* d_ws, size_t ws_size,
                              hipStream_t stream) {
    const float* logits     = (const float*)d_in[0];  // (N, 81)
    const float* regression = (const float*)d_in[1];  // (N, 405)
    const float* boxes      = (const float*)d_in[2];  // (N, 5)
    float*       out        = (float*)d_out;          // (N, 486)

    const int n = in_sizes[0] / CNUM;                 // N = 200000
    const int blocks = (n + WAVES_PB - 1) / WAVES_PB; // 25000
    PostProcessor_54374285967910_kernel<<<blocks, 256, 0, stream>>>(
        logits, regression, boxes, out, n);
}